// GraphConvolutionNetwork_72688026518111
// MI455X (gfx1250) — compile-verified
//
#include <hip/hip_runtime.h>
#include <hip/hip_bf16.h>

#define HID   128
#define GMAX  100
#define CHUNK 256
#define EPSV  1e-5f
#define WSTRIDE 144   // float2 stride per K-pair row; 144 % 32 == 16 -> conflict-free b64

typedef __attribute__((ext_vector_type(2))) float v2f;
typedef __attribute__((ext_vector_type(8))) float v8f;

// ---------------------------------------------------------------------------
// GEMM: H[N,128] = X[N,128] @ W[128,128], fp32 via V_WMMA_F32_16X16X4_F32.
// Block = 256 threads = 8 waves; each wave computes a 16-row x 128-col tile.
// W staged in LDS pre-paired by K: Ws2[k/2][n] = (W[k][n], W[k+1][n]) so each
// B fragment is a single ds_load_b64 (no register repacking).
// Safe for X == H (each wave reads only its own 16 rows, stores afterwards).
// ---------------------------------------------------------------------------
__global__ void gcn_gemm_wmma(const float* X, const float* __restrict__ Wl,
                              float* H, int N) {
  __shared__ v2f Ws2[64 * WSTRIDE];
  const int tid = threadIdx.x;
  for (int i = tid; i < HID * HID / 2; i += 256) {
    const int pr = i >> 7;        // K-pair row (k/2)
    const int n  = i & 127;       // column
    v2f w;
    w.x = Wl[(2 * pr) * HID + n];
    w.y = Wl[(2 * pr + 1) * HID + n];
    Ws2[pr * WSTRIDE + n] = w;
  }
  __syncthreads();

  const int wave = tid >> 5;
  const int lane = tid & 31;
  const int half = lane >> 4;     // 0: lanes 0-15, 1: lanes 16-31
  const int l16  = lane & 15;
  const long tile = (long)blockIdx.x * 8 + wave;
  const long rowBase = tile * 16;
  if (rowBase >= N) return;       // wave-uniform: EXEC stays all-ones inside

  v8f acc[8] = {};
  long arow = rowBase + l16;
  if (arow >= N) arow = N - 1;    // clamp reads for ragged tail (stores guarded)
  const float* ap = X + arow * HID;

  for (int k = 0; k < HID; k += 4) {
    const int ka = k + half * 2;  // lane-half selects K / K+2 per ISA layout
    v2f a;
    a.x = ap[ka];
    a.y = ap[ka + 1];
    const v2f* bp = &Ws2[((k >> 1) + half) * WSTRIDE + l16];
#pragma unroll
    for (int j = 0; j < 8; ++j) {
      const v2f bf = bp[j * 16];  // single ds_load_b64, immediate offset j*128B
      acc[j] = __builtin_amdgcn_wmma_f32_16x16x4_f32(
          false, a, false, bf, (short)0, acc[j], false, false);
    }
  }

  // C/D layout: vgpr r, lanes 0-15 -> row rowBase+r, lanes 16-31 -> +8
#pragma unroll
  for (int j = 0; j < 8; ++j) {
#pragma unroll
    for (int r = 0; r < 8; ++r) {
      const long row = rowBase + r + 8 * half;
      if (row < N) H[row * HID + j * 16 + l16] = acc[j][r];
    }
  }
}

// ---------------------------------------------------------------------------
// Degree / normalization
// ---------------------------------------------------------------------------
__global__ void gcn_deg_init(float* deg, int N) {
  const long i = (long)blockIdx.x * blockDim.x + threadIdx.x;
  if (i < N) deg[i] = 1.0f;  // self-loop weight
}

__global__ void gcn_deg_edge(const int* __restrict__ dst,
                             const float* __restrict__ ew, float* deg, int E) {
  const long e = (long)blockIdx.x * blockDim.x + threadIdx.x;
  if (e < E) atomicAdd(&deg[dst[e]], ew[e]);
}

__global__ void gcn_deg_finish(float* deg, int N) {  // deg -> dis in place
  const long i = (long)blockIdx.x * blockDim.x + threadIdx.x;
  if (i < N) {
    const float d = deg[i];
    deg[i] = d > 0.0f ? rsqrtf(d) : 0.0f;
  }
}

// counts[g] = #nodes in graph g (batch sorted -> flush trick)
__global__ void gcn_count_nodes(const int* __restrict__ batch, float* cnt, int N) {
  if (threadIdx.x != 0) return;
  long n0 = (long)blockIdx.x * 1024;
  long n1 = n0 + 1024; if (n1 > N) n1 = N;
  if (n0 >= n1) return;
  int curg = batch[n0];
  float acc = 0.0f;
  for (long n = n0; n < n1; ++n) {
    const int g = batch[n];
    if (g != curg) { atomicAdd(&cnt[curg], acc); acc = 0.0f; curg = g; }
    acc += 1.0f;
  }
  atomicAdd(&cnt[curg], acc);
}

// ---------------------------------------------------------------------------
// agg[n,c] = b[c] + dis[n]^2 * h[n,c]   (self-loop contribution folded in)
// block = 256 threads -> 8 rows, float4 per lane
// ---------------------------------------------------------------------------
__global__ void gcn_agg_init(const float* __restrict__ H,
                             const float* __restrict__ dis,
                             const float* __restrict__ bias,
                             float* __restrict__ agg, int N) {
  const long row = (long)blockIdx.x * 8 + (threadIdx.x >> 5);
  if (row >= N) return;
  const int c4 = (threadIdx.x & 31) * 4;
  const float d = dis[row];
  const float dd = d * d;
  const float4 h = *(const float4*)(H + row * HID + c4);
  const float4 bb = *(const float4*)(bias + c4);
  float4 o;
  o.x = bb.x + dd * h.x;
  o.y = bb.y + dd * h.y;
  o.z = bb.z + dd * h.z;
  o.w = bb.w + dd * h.w;
  *(float4*)(agg + row * HID + c4) = o;
}

// ---------------------------------------------------------------------------
// Edge scatter: one wave per edge; lane handles 4 channels (float4 gather,
// 4x global_atomic_add_f32, fire-and-forget).
// ---------------------------------------------------------------------------
__global__ void gcn_edge_scatter(const int* __restrict__ src,
                                 const int* __restrict__ dst,
                                 const float* __restrict__ ew,
                                 const float* __restrict__ dis,
                                 const float* __restrict__ H,
                                 float* agg, int E) {
  const int lane = threadIdx.x & 31;
  const long e = (long)blockIdx.x * 8 + (threadIdx.x >> 5);
  if (e >= E) return;
  const int s = src[e];
  const int t = dst[e];
  const float coeff = dis[s] * ew[e] * dis[t];
  const float4 v = *((const float4*)(H + (long)s * HID) + lane);
  float* ap = agg + (long)t * HID + lane * 4;
  atomicAdd(ap + 0, coeff * v.x);
  atomicAdd(ap + 1, coeff * v.y);
  atomicAdd(ap + 2, coeff * v.z);
  atomicAdd(ap + 3, coeff * v.w);
}

// ---------------------------------------------------------------------------
// Per-graph channel sums over sorted batch: block = 128 threads (one/channel),
// scans CHUNK nodes, flushes register accumulator at graph boundaries only.
// ---------------------------------------------------------------------------
__global__ void gcn_graph_sum(const float* __restrict__ X,
                              const int* __restrict__ batch,
                              float* gsum, int N) {
  const int c = threadIdx.x;
  long n0 = (long)blockIdx.x * CHUNK;
  long n1 = n0 + CHUNK; if (n1 > N) n1 = N;
  if (n0 >= n1) return;
  int curg = batch[n0];
  float acc = 0.0f;
  for (long n = n0; n < n1; ++n) {
    const int g = batch[n];
    if (g != curg) { atomicAdd(&gsum[(long)curg * HID + c], acc); acc = 0.0f; curg = g; }
    acc += X[n * HID + c];
  }
  atomicAdd(&gsum[(long)curg * HID + c], acc);
}

// sub = agg - alpha*mean  (in place), accumulate per-graph sum of sub^2
__global__ void gcn_sub_var(float* X, const int* __restrict__ batch,
                            const float* __restrict__ gsum,
                            const float* __restrict__ cnt,
                            const float* __restrict__ alpha,
                            float* vsum, int N) {
  const int c = threadIdx.x;
  long n0 = (long)blockIdx.x * CHUNK;
  long n1 = n0 + CHUNK; if (n1 > N) n1 = N;
  if (n0 >= n1) return;
  const float al = alpha[c];
  int curg = batch[n0];
  float mean = gsum[(long)curg * HID + c] / cnt[curg];
  float acc = 0.0f;
  for (long n = n0; n < n1; ++n) {
    const int g = batch[n];
    if (g != curg) {
      atomicAdd(&vsum[(long)curg * HID + c], acc);
      acc = 0.0f; curg = g;
      mean = gsum[(long)curg * HID + c] / cnt[curg];
    }
    const float sub = X[n * HID + c] - al * mean;
    X[n * HID + c] = sub;
    acc += sub * sub;
  }
  atomicAdd(&vsum[(long)curg * HID + c], acc);
}

// out = relu(gw * sub * rsqrt(var + eps) + gb)
__global__ void gcn_norm_final(const float* Sub, const int* __restrict__ batch,
                               const float* __restrict__ vsum,
                               const float* __restrict__ cnt,
                               const float* __restrict__ gw,
                               const float* __restrict__ gb,
                               float* Out, int N) {
  const long row = (long)blockIdx.x * 8 + (threadIdx.x >> 5);
  if (row >= N) return;
  const int c4 = (threadIdx.x & 31) * 4;
  const int g = batch[row];
  const float ic = 1.0f / cnt[g];
  const float4 s = *(const float4*)(Sub + row * HID + c4);
  const float4 v = *(const float4*)(vsum + (long)g * HID + c4);
  const float4 w = *(const float4*)(gw + c4);
  const float4 b = *(const float4*)(gb + c4);
  float4 o;
  o.x = fmaxf(w.x * s.x * rsqrtf(v.x * ic + EPSV) + b.x, 0.0f);
  o.y = fmaxf(w.y * s.y * rsqrtf(v.y * ic + EPSV) + b.y, 0.0f);
  o.z = fmaxf(w.z * s.z * rsqrtf(v.z * ic + EPSV) + b.z, 0.0f);
  o.w = fmaxf(w.w * s.w * rsqrtf(v.w * ic + EPSV) + b.w, 0.0f);
  *(float4*)(Out + row * HID + c4) = o;
}

// ---------------------------------------------------------------------------
extern "C" void kernel_launch(void* const* d_in, const int* in_sizes, int n_in,
                              void* d_out, int out_size, void* d_ws, size_t ws_size,
                              hipStream_t stream) {
  const float* node  = (const float*)d_in[0];
  const int*   ei    = (const int*)d_in[1];
  const float* ew    = (const float*)d_in[2];
  const int*   batch = (const int*)d_in[3];
  const float* W     = (const float*)d_in[4];
  const float* b     = (const float*)d_in[5];
  const float* gw    = (const float*)d_in[6];
  const float* gb    = (const float*)d_in[7];
  const float* ga    = (const float*)d_in[8];
  float* out = (float*)d_out;

  const int N = in_sizes[0] / HID;
  const int E = in_sizes[1] / 2;
  const int L = in_sizes[4] / (HID * HID);
  const int* src = ei;
  const int* dst = ei + E;

  // workspace layout (floats): A[N*128] | dis[N] | gsum[G*128] | vsum[G*128] | cnt[G]
  float* A    = (float*)d_ws;
  float* dis  = A + (size_t)N * HID;
  float* gsum = dis + N;
  float* vsum = gsum + (size_t)GMAX * HID;
  float* cnt  = vsum + (size_t)GMAX * HID;

  // ---- once: degrees -> dis, per-graph node counts ----
  gcn_deg_init<<<(N + 255) / 256, 256, 0, stream>>>(dis, N);
  gcn_deg_edge<<<(E + 255) / 256, 256, 0, stream>>>(dst, ew, dis, E);
  gcn_deg_finish<<<(N + 255) / 256, 256, 0, stream>>>(dis, N);
  hipMemsetAsync(cnt, 0, GMAX * sizeof(float), stream);
  gcn_count_nodes<<<(N + 1023) / 1024, 32, 0, stream>>>(batch, cnt, N);

  const int rowTiles    = (N + 15) / 16;
  const int gemmBlocks  = (rowTiles + 7) / 8;
  const int rows8       = (N + 7) / 8;
  const int chunkBlocks = (N + CHUNK - 1) / CHUNK;
  const int edgeBlocks  = (E + 7) / 8;

  const float* X = node;
  for (int l = 0; l < L; ++l) {
    const float* Wl = W + (size_t)l * HID * HID;
    const float* bl = b + (size_t)l * HID;

    gcn_gemm_wmma<<<gemmBlocks, 256, 0, stream>>>(X, Wl, A, N);
    gcn_agg_init<<<rows8, 256, 0, stream>>>(A, dis, bl, out, N);
    gcn_edge_scatter<<<edgeBlocks, 256, 0, stream>>>(src, dst, ew, dis, A, out, E);

    hipMemsetAsync(gsum, 0, (size_t)GMAX * HID * sizeof(float), stream);
    hipMemsetAsync(vsum, 0, (size_t)GMAX * HID * sizeof(float), stream);
    gcn_graph_sum<<<chunkBlocks, HID, 0, stream>>>(out, batch, gsum, N);
    gcn_sub_var<<<chunkBlocks, HID, 0, stream>>>(out, batch, gsum, cnt,
                                                 ga + (size_t)l * HID, vsum, N);

    float* layerOut = (l == L - 1) ? out : A;  // last layer lands in d_out
    gcn_norm_final<<<rows8, 256, 0, stream>>>(out, batch, vsum, cnt,
                                              gw + (size_t)l * HID,
                                              gb + (size_t)l * HID, layerOut, N);
    X = A;  // next layer reads A (gemm is in-place safe)
  }
}